// MultimodalRNN_9337258901850
// MI455X (gfx1250) — compile-verified
//
#include <hip/hip_runtime.h>
#include <hip/hip_bf16.h>
#include <math.h>

// ---------------------------------------------------------------------------
// Types for WMMA f32 16x16x4: A is 16x4 f32 (2 VGPRs/lane), B is 4x16 f32
// (2 VGPRs/lane), C/D is 16x16 f32 (8 VGPRs/lane).
// ---------------------------------------------------------------------------
typedef float v2f __attribute__((ext_vector_type(2)));
typedef float v8f __attribute__((ext_vector_type(8)));

#define E_DIM 1024
#define H_DIM 2048
#define G_DIM (4 * H_DIM)   // 8192 gate width

// ---------------------------------------------------------------------------
// Kernel 1: gates_x[t, :] = emb[idx[t], :] @ W_ih^T   via V_WMMA_F32_16X16X4_F32
// One wave (32 threads) per block; each block computes a 16(M) x 64(N) tile,
// reusing the A fragment across 4 accumulators.
// grid = (N/64, T/16), block = 32.
//
// A fragment layout (32-bit A 16x4): lane L -> m = L&15, kh = L>>4;
//   a.x = A[m][2*kh], a.y = A[m][2*kh+1]           (contiguous float2 load)
// B fragment layout (32-bit B 4x16): lane L -> n = L&15, kh = L>>4;
//   b.x = B[2*kh][n], b.y = B[2*kh+1][n]  with B[k][n] = W_ih[n][k] (row-major
//   W_ih, so this is also a contiguous float2 along E).
// D layout: VGPR r -> row (r + 8*kh), col (L&15).
// ---------------------------------------------------------------------------
__global__ void __launch_bounds__(32)
gemm_gather_wmma(const int* __restrict__ idx,
                 const float* __restrict__ emb,   // [V, E]
                 const float* __restrict__ Wih,   // [G, E] row-major
                 float* __restrict__ gx)          // [T, G]
{
    const int lane = threadIdx.x & 31;
    const int m    = lane & 15;
    const int kh   = lane >> 4;            // 0 or 1
    const int trow = blockIdx.y * 16 + m;  // A row (time step)
    const int n0   = blockIdx.x * 64;      // N tile base
    const int nb   = n0 + m;               // B lane column base

    const float* arow = emb + (size_t)idx[trow] * E_DIM;
    const float* b0r  = Wih + (size_t)(nb +  0) * E_DIM;
    const float* b1r  = Wih + (size_t)(nb + 16) * E_DIM;
    const float* b2r  = Wih + (size_t)(nb + 32) * E_DIM;
    const float* b3r  = Wih + (size_t)(nb + 48) * E_DIM;

    v8f acc0 = {}, acc1 = {}, acc2 = {}, acc3 = {};

    for (int k0 = 0; k0 < E_DIM; k0 += 4) {
        const int k = k0 + 2 * kh;
        v2f a  = *(const v2f*)(arow + k);
        v2f b0 = *(const v2f*)(b0r + k);
        v2f b1 = *(const v2f*)(b1r + k);
        v2f b2 = *(const v2f*)(b2r + k);
        v2f b3 = *(const v2f*)(b3r + k);
        acc0 = __builtin_amdgcn_wmma_f32_16x16x4_f32(false, a, false, b0, (short)0, acc0, false, false);
        acc1 = __builtin_amdgcn_wmma_f32_16x16x4_f32(false, a, false, b1, (short)0, acc1, false, false);
        acc2 = __builtin_amdgcn_wmma_f32_16x16x4_f32(false, a, false, b2, (short)0, acc2, false, false);
        acc3 = __builtin_amdgcn_wmma_f32_16x16x4_f32(false, a, false, b3, (short)0, acc3, false, false);
    }

    const int nd = n0 + (lane & 15);
    const int md = blockIdx.y * 16 + 8 * kh;
#pragma unroll
    for (int r = 0; r < 8; ++r) {
        float* orow = gx + (size_t)(md + r) * G_DIM + nd;
        orow[0]  = acc0[r];
        orow[16] = acc1[r];
        orow[32] = acc2[r];
        orow[48] = acc3[r];
    }
}

// ---------------------------------------------------------------------------
// Kernel 2: one LSTM recurrence step (launched T times; W_hh stays L2-resident).
// 256 threads = 8 waves; one wave per hidden unit u (8 units/block, 256 blocks).
// Each wave computes the 4 gate dot products dot(h, W_hh[g*H+u, :]) with
// float4 loads and a wave32 shuffle reduction; h is staged in LDS.
// ---------------------------------------------------------------------------
__global__ void __launch_bounds__(256)
lstm_step(const float* __restrict__ gx_t,   // [G] input-projection for step t
          const float* __restrict__ b_ih,   // [G]
          const float* __restrict__ b_hh,   // [G]
          const float* __restrict__ Whh,    // [G, H] row-major
          const float* __restrict__ h_in,   // [H]
          float* __restrict__ h_out,        // [H]
          float* __restrict__ c)            // [H] (in-place)
{
    __shared__ float hs[H_DIM];
    const int tid = threadIdx.x;
    for (int i = tid; i < H_DIM; i += 256) hs[i] = h_in[i];
    __syncthreads();

    const int wave = tid >> 5;
    const int lane = tid & 31;
    const int u    = blockIdx.x * 8 + wave;

    float dot[4];
#pragma unroll
    for (int g = 0; g < 4; ++g) {
        const float* wrow = Whh + (size_t)(g * H_DIM + u) * H_DIM;
        float s = 0.f;
        for (int k = lane * 4; k < H_DIM; k += 32 * 4) {
            float4 w = *(const float4*)(wrow + k);
            s += w.x * hs[k] + w.y * hs[k + 1] + w.z * hs[k + 2] + w.w * hs[k + 3];
        }
#pragma unroll
        for (int off = 16; off > 0; off >>= 1) s += __shfl_xor(s, off, 32);
        dot[g] = s;
    }

    if (lane == 0) {
        const float pi = gx_t[0 * H_DIM + u] + b_ih[0 * H_DIM + u] + b_hh[0 * H_DIM + u] + dot[0];
        const float pf = gx_t[1 * H_DIM + u] + b_ih[1 * H_DIM + u] + b_hh[1 * H_DIM + u] + dot[1];
        const float pg = gx_t[2 * H_DIM + u] + b_ih[2 * H_DIM + u] + b_hh[2 * H_DIM + u] + dot[2];
        const float po = gx_t[3 * H_DIM + u] + b_ih[3 * H_DIM + u] + b_hh[3 * H_DIM + u] + dot[3];
        const float ig = 1.f / (1.f + expf(-pi));
        const float fg = 1.f / (1.f + expf(-pf));
        const float gg = tanhf(pg);
        const float og = 1.f / (1.f + expf(-po));
        const float cn = fg * c[u] + ig * gg;
        c[u]     = cn;
        h_out[u] = og * tanhf(cn);
    }
}

// ---------------------------------------------------------------------------
// Kernel 3: GEMV y = act(W @ concat(x1[K1], x2[K2]) + b). One wave per row.
// ---------------------------------------------------------------------------
__global__ void __launch_bounds__(256)
gemv_act(const float* __restrict__ W,   // [M, K1+K2] row-major
         const float* __restrict__ b,   // [M]
         const float* __restrict__ x1, int K1,
         const float* __restrict__ x2, int K2,
         float* __restrict__ y, int M, int do_relu)
{
    const int wave = threadIdx.x >> 5;
    const int lane = threadIdx.x & 31;
    const int row  = blockIdx.x * 8 + wave;
    if (row >= M) return;

    const float* wr = W + (size_t)row * (K1 + K2);
    float s = 0.f;
    for (int k = lane * 4; k < K1; k += 32 * 4) {
        float4 w = *(const float4*)(wr + k);
        s += w.x * x1[k] + w.y * x1[k + 1] + w.z * x1[k + 2] + w.w * x1[k + 3];
    }
    for (int k = lane * 4; k < K2; k += 32 * 4) {
        float4 w = *(const float4*)(wr + K1 + k);
        s += w.x * x2[k] + w.y * x2[k + 1] + w.z * x2[k + 2] + w.w * x2[k + 3];
    }
#pragma unroll
    for (int off = 16; off > 0; off >>= 1) s += __shfl_xor(s, off, 32);

    if (lane == 0) {
        s += b[row];
        y[row] = do_relu ? fmaxf(s, 0.f) : s;
    }
}

__global__ void zero_kernel(float* __restrict__ p, int n)
{
    int i = blockIdx.x * blockDim.x + threadIdx.x;
    if (i < n) p[i] = 0.f;
}

// ---------------------------------------------------------------------------
extern "C" void kernel_launch(void* const* d_in, const int* in_sizes, int n_in,
                              void* d_out, int out_size, void* d_ws, size_t ws_size,
                              hipStream_t stream) {
    (void)in_sizes; (void)n_in; (void)out_size; (void)ws_size;

    const int*   x_title   = (const int*)  d_in[0];   // [128]
    const int*   x_authors = (const int*)  d_in[1];   // [16]
    const float* emb_t     = (const float*)d_in[2];   // [50000,1024]
    const float* emb_a     = (const float*)d_in[3];   // [20000,1024]
    const float* W_ih_t    = (const float*)d_in[4];   // [8192,1024]
    const float* W_hh_t    = (const float*)d_in[5];   // [8192,2048]
    const float* b_ih_t    = (const float*)d_in[6];
    const float* b_hh_t    = (const float*)d_in[7];
    const float* W_ih_a    = (const float*)d_in[8];
    const float* W_hh_a    = (const float*)d_in[9];
    const float* b_ih_a    = (const float*)d_in[10];
    const float* b_hh_a    = (const float*)d_in[11];
    const float* W1        = (const float*)d_in[12];  // [2048,4096]
    const float* b1        = (const float*)d_in[13];
    const float* W2        = (const float*)d_in[14];  // [2048,2048]
    const float* b2        = (const float*)d_in[15];
    const float* W3        = (const float*)d_in[16];  // [2,2048]
    const float* b3        = (const float*)d_in[17];

    const int TT = 128, TA = 16;

    // Workspace carve-up (floats)
    float* ws   = (float*)d_ws;
    float* gx_t = ws;                     ws += (size_t)TT * G_DIM;  // 4 MB
    float* gx_a = ws;                     ws += (size_t)TA * G_DIM;  // 0.5 MB
    float* hA_t = ws;                     ws += H_DIM;
    float* hB_t = ws;                     ws += H_DIM;
    float* c_t  = ws;                     ws += H_DIM;
    float* hA_a = ws;                     ws += H_DIM;
    float* hB_a = ws;                     ws += H_DIM;
    float* c_a  = ws;                     ws += H_DIM;
    float* h1   = ws;                     ws += H_DIM;
    float* h2   = ws;                     ws += H_DIM;

    // Zero initial h / c state (hA_t..c_a is contiguous: 6*H floats)
    {
        const int n = 6 * H_DIM;
        zero_kernel<<<(n + 255) / 256, 256, 0, stream>>>(hA_t, n);
    }

    // Input projections (WMMA f32): gx = gather(emb, idx) @ W_ih^T
    {
        dim3 grid_t(G_DIM / 64, TT / 16);     // (128, 8)
        gemm_gather_wmma<<<grid_t, 32, 0, stream>>>(x_title, emb_t, W_ih_t, gx_t);
        dim3 grid_a(G_DIM / 64, TA / 16);     // (128, 1)
        gemm_gather_wmma<<<grid_a, 32, 0, stream>>>(x_authors, emb_a, W_ih_a, gx_a);
    }

    // Title LSTM recurrence (W_hh_t is 64 MB -> L2 resident across steps)
    float* hin  = hA_t;
    float* hout = hB_t;
    for (int t = 0; t < TT; ++t) {
        lstm_step<<<H_DIM / 8, 256, 0, stream>>>(gx_t + (size_t)t * G_DIM,
                                                 b_ih_t, b_hh_t, W_hh_t,
                                                 hin, hout, c_t);
        float* tmp = hin; hin = hout; hout = tmp;
    }
    const float* hT_final = hin;

    // Authors LSTM recurrence
    hin = hA_a; hout = hB_a;
    for (int t = 0; t < TA; ++t) {
        lstm_step<<<H_DIM / 8, 256, 0, stream>>>(gx_a + (size_t)t * G_DIM,
                                                 b_ih_a, b_hh_a, W_hh_a,
                                                 hin, hout, c_a);
        float* tmp = hin; hin = hout; hout = tmp;
    }
    const float* hA_final = hin;

    // MLP head
    gemv_act<<<H_DIM / 8, 256, 0, stream>>>(W1, b1, hT_final, H_DIM, hA_final, H_DIM,
                                            h1, H_DIM, 1);
    gemv_act<<<H_DIM / 8, 256, 0, stream>>>(W2, b2, h1, H_DIM, h1, 0,
                                            h2, H_DIM, 1);
    gemv_act<<<1, 256, 0, stream>>>(W3, b3, h2, H_DIM, h2, 0,
                                    (float*)d_out, 2, 0);
}